// GraphConvolutionLayer_GraphSAGE_45423574123238
// MI455X (gfx1250) — compile-verified
//
#include <hip/hip_runtime.h>

typedef float v2f __attribute__((ext_vector_type(2)));
typedef float v8f __attribute__((ext_vector_type(8)));

#define DIM 64   // both input dim D and latent dim L are 64

// ---------------------------------------------------------------------------
// Kernel 1: pooled = x   (residual folded into the accumulator buffer)
// ---------------------------------------------------------------------------
__global__ void init_pooled(const float* __restrict__ x,
                            float* __restrict__ pooled, int n4) {
  int i = blockIdx.x * blockDim.x + threadIdx.x;
  if (i < n4) {
    ((float4*)pooled)[i] = ((const float4*)x)[i];
  }
}

// ---------------------------------------------------------------------------
// Kernel 2: scatter-add x[src] rows into pooled[dst].
// One wave (32 lanes) per edge; each lane handles 2 consecutive columns.
// x and pooled are L2-resident (12.8 MB each vs 192 MB L2).
// ---------------------------------------------------------------------------
__global__ void edge_scatter(const float* __restrict__ x,
                             const int* __restrict__ esrc,
                             const int* __restrict__ edst,
                             float* __restrict__ pooled, int E) {
  int gid = blockIdx.x * blockDim.x + threadIdx.x;
  int e = gid >> 5;
  if (e >= E) return;
  int c = (gid & 31) << 1;
  int s = esrc[e];
  int d = edst[e];
  float2 v = *(const float2*)(x + s * DIM + c);
  float* dp = pooled + d * DIM + c;
  unsafeAtomicAdd(dp,     v.x);   // global_atomic_add_f32
  unsafeAtomicAdd(dp + 1, v.y);
}

// ---------------------------------------------------------------------------
// Kernel 3: out = relu((pooled @ weight + bias) / degree)
// 8 waves / block, each wave owns a 16-row tile of the output [16 x 64].
// Full-precision fp32 WMMA: V_WMMA_F32_16X16X4_F32, K=64 in 16 steps of 4.
//
// Weight is staged in LDS pre-packed into B-fragment layout so each lane's
// fragment is a single contiguous ds_load_b64 (even-aligned VGPR pair, no
// repack movs) and a wave's loads cover 256 consecutive bytes -> all 64 LDS
// banks exactly once (conflict-free).
//
// A fragment layout (16x4 f32): lanes 0-15 -> row=lane,    K = k,k+1
//                               lanes 16-31 -> row=lane-16, K = k+2,k+3
// B fragment layout (4x16 f32): lanes 0-15 -> col=lane,    K = k,k+1
//                               lanes 16-31 -> col=lane-16, K = k+2,k+3
// C/D layout: VGPR j: lanes 0-15 -> M=j, lanes 16-31 -> M=j+8; N = lane&15
// ---------------------------------------------------------------------------
__global__ void __launch_bounds__(256)
gemm_norm_relu(const float* __restrict__ pooled,
               const float* __restrict__ weight,
               const float* __restrict__ bias,
               const float* __restrict__ degree,
               float* __restrict__ out, int ntiles, int nrows) {
  // wfrag[kstep][nblk][lane] = { W[krow][col], W[krow+1][col] }
  //   krow = 4*kstep + 2*(lane>>4),  col = 16*nblk + (lane&15)
  // 16 ksteps * 4 nblks * 32 lanes * 2 floats = 4096 floats = 16 KB
  __shared__ float wfrag[16 * 4 * 32 * 2];

  for (int p = threadIdx.x; p < 16 * 4 * 32; p += 256) {
    int l  = p & 31;          // lane
    int nb = (p >> 5) & 3;    // n-block
    int ks = p >> 7;          // k-step
    int col  = nb * 16 + (l & 15);
    int krow = ks * 4 + ((l >> 4) << 1);
    wfrag[p * 2]     = weight[krow * DIM + col];
    wfrag[p * 2 + 1] = weight[(krow + 1) * DIM + col];
  }
  __syncthreads();

  const int wave = threadIdx.x >> 5;
  const int lane = threadIdx.x & 31;
  const int tile = blockIdx.x * 8 + wave;
  if (tile >= ntiles) return;              // wave-uniform branch

  const int hi   = lane >> 4;              // 0: lanes 0-15, 1: lanes 16-31
  const int l15  = lane & 15;
  const int kOff = hi << 1;                // 0 or 2
  const int rowBase = tile << 4;

  // Clamp keeps A loads in-bounds on a ragged tail without divergence.
  int aRow = rowBase + l15;
  aRow = aRow < nrows ? aRow : (nrows - 1);
  const float* aPtr = pooled + aRow * DIM + kOff;

  const v2f* bfrag = (const v2f*)wfrag + lane;   // index by (kstep*4+nb)*32

  v8f acc0 = {}, acc1 = {}, acc2 = {}, acc3 = {};

#pragma unroll
  for (int ks = 0; ks < 16; ++ks) {
    v2f a;
    a.x = aPtr[ks * 4];
    a.y = aPtr[ks * 4 + 1];

    v2f b0 = bfrag[(ks * 4 + 0) * 32];   // single ds_load_b64 each
    v2f b1 = bfrag[(ks * 4 + 1) * 32];
    v2f b2 = bfrag[(ks * 4 + 2) * 32];
    v2f b3 = bfrag[(ks * 4 + 3) * 32];

    acc0 = __builtin_amdgcn_wmma_f32_16x16x4_f32(false, a, false, b0, (short)0, acc0, false, false);
    acc1 = __builtin_amdgcn_wmma_f32_16x16x4_f32(false, a, false, b1, (short)0, acc1, false, false);
    acc2 = __builtin_amdgcn_wmma_f32_16x16x4_f32(false, a, false, b2, (short)0, acc2, false, false);
    acc3 = __builtin_amdgcn_wmma_f32_16x16x4_f32(false, a, false, b3, (short)0, acc3, false, false);
  }

  // Epilogue: +bias, exact fp32 /degree, ReLU, store.
  const float bb0 = bias[l15];
  const float bb1 = bias[16 + l15];
  const float bb2 = bias[32 + l15];
  const float bb3 = bias[48 + l15];

#pragma unroll
  for (int j = 0; j < 8; ++j) {
    int row = rowBase + (hi << 3) + j;
    if (row < nrows) {
      float dg = degree[row];
      float* orow = out + row * DIM + l15;
      float v0 = (acc0[j] + bb0) / dg;
      float v1 = (acc1[j] + bb1) / dg;
      float v2 = (acc2[j] + bb2) / dg;
      float v3 = (acc3[j] + bb3) / dg;
      orow[0]  = v0 > 0.0f ? v0 : 0.0f;
      orow[16] = v1 > 0.0f ? v1 : 0.0f;
      orow[32] = v2 > 0.0f ? v2 : 0.0f;
      orow[48] = v3 > 0.0f ? v3 : 0.0f;
    }
  }
}

// ---------------------------------------------------------------------------
// Inputs (setup_inputs order): x[N*64], weight[64*64], bias[64],
//                              node_degree[N], edge_src[E], edge_dst[E]
// Output: relu(((segment_sum + x) @ W + b) / degree)  -> float32 [N*64]
// Workspace: pooled accumulator, N*64 floats (12.8 MB)
// ---------------------------------------------------------------------------
extern "C" void kernel_launch(void* const* d_in, const int* in_sizes, int n_in,
                              void* d_out, int out_size, void* d_ws, size_t ws_size,
                              hipStream_t stream) {
  const float* x      = (const float*)d_in[0];
  const float* weight = (const float*)d_in[1];
  const float* bias   = (const float*)d_in[2];
  const float* degree = (const float*)d_in[3];
  const int*   esrc   = (const int*)d_in[4];
  const int*   edst   = (const int*)d_in[5];
  float* out = (float*)d_out;

  const int N = in_sizes[3];   // node_degree has N elements
  const int E = in_sizes[4];

  float* pooled = (float*)d_ws;

  // 1) pooled = x
  int n4 = (N * DIM) / 4;
  init_pooled<<<(n4 + 255) / 256, 256, 0, stream>>>(x, pooled, n4);

  // 2) pooled[dst] += x[src]  (one wave per edge)
  long long scatterThreads = (long long)E * 32;
  int scatterBlocks = (int)((scatterThreads + 255) / 256);
  edge_scatter<<<scatterBlocks, 256, 0, stream>>>(x, esrc, edst, pooled, E);

  // 3) fused GEMM + bias + degree-normalize + relu via fp32 WMMA
  int ntiles = (N + 15) / 16;
  gemm_norm_relu<<<(ntiles + 7) / 8, 256, 0, stream>>>(pooled, weight, bias,
                                                       degree, out, ntiles, N);
}